// GAM_8246337208834
// MI455X (gfx1250) — compile-verified
//
#include <hip/hip_runtime.h>

typedef __attribute__((ext_vector_type(2))) float v2f;
typedef __attribute__((ext_vector_type(8))) float v8f;

#define C_CH   128
#define P_TOT  16384   // H*W = 128*128
#define TILE_P 128
#define ROWSTR 132     // padded row stride (floats): 132*4 = 528 B, 16B-aligned

// out[b,c,p] = x1[b,c,p] + (gamma/C) * sum_c' x1[b,c',p]
// (reference's softmax over a j-constant energy is exactly uniform 1/C)
__global__ __launch_bounds__(256)
void gam_fused_kernel(const float* __restrict__ x1,
                      const float* __restrict__ gamma,
                      float* __restrict__ out)
{
    __shared__ __align__(16) float tile[C_CH * ROWSTR]; // ~66 KB
    __shared__ __align__(16) float ssum[TILE_P];

    const int t      = threadIdx.x;
    const int b      = blockIdx.x >> 7;              // 128 p-tiles per batch
    const int tile_p = (blockIdx.x & 127) * TILE_P;
    const int lane   = t & 31;
    const int wave   = t >> 5;

    // ---- Stage 128ch x 128p fp32 tile: async global -> LDS (b128 chunks) ----
    const int chunk = t & 31;   // 16B chunk index within a 512B row (32 chunks)
    const int crow0 = t >> 5;   // 8 channel rows covered per iteration
    #pragma unroll
    for (int i = 0; i < 16; ++i) {
        const int c = crow0 + 8 * i;
        const float* gsrc =
            x1 + ((size_t)(b * C_CH + c) * P_TOT + tile_p + chunk * 4);
        // low 32 bits of a flat LDS pointer == wave-relative LDS byte offset
        unsigned lds_off = (unsigned)(size_t)(&tile[c * ROWSTR + chunk * 4]);
        unsigned long long gaddr = (unsigned long long)gsrc;
        asm volatile("global_load_async_to_lds_b128 %0, %1, off"
                     :: "v"(lds_off), "v"(gaddr) : "memory");
    }
    asm volatile("s_wait_asynccnt 0x0" ::: "memory");
    __syncthreads();

    // ---- Phase 1: channel sum via V_WMMA_F32_16X16X4_F32 (ones x tile) ----
    // Each wave reduces a 16-column p-strip; 32 accumulating WMMAs cover C=128.
    const int pcol = (wave << 4) + (lane & 15);
    v2f aones; aones.x = 1.0f; aones.y = 1.0f;
    v8f acc = {};
    const int cbase0 = (lane < 16) ? 0 : 2;
    #pragma unroll 4
    for (int k = 0; k < 32; ++k) {
        const int c0 = 4 * k + cbase0;
        v2f bmat;
        bmat.x = tile[(c0    ) * ROWSTR + pcol];
        bmat.y = tile[(c0 + 1) * ROWSTR + pcol];
        acc = __builtin_amdgcn_wmma_f32_16x16x4_f32(
            /*neg_a=*/false, aones, /*neg_b=*/false, bmat,
            /*c_mod=*/(short)0, acc, /*reuse_a=*/false, /*reuse_b=*/false);
    }
    // every D element equals the column sum; take row M=0 from lanes 0..15
    if (lane < 16) ssum[(wave << 4) + lane] = acc[0];
    __syncthreads();

    // ---- Phase 2: out = x1 + (gamma/C)*colsum, float4 in/out ----
    const float scale = gamma[0] * (1.0f / (float)C_CH);
    const float4 s4 = *(const float4*)&ssum[chunk * 4];
    #pragma unroll
    for (int i = 0; i < 16; ++i) {
        const int c = crow0 + 8 * i;
        const float4 x4 = *(const float4*)&tile[c * ROWSTR + chunk * 4];
        float4 o;
        o.x = fmaf(scale, s4.x, x4.x);
        o.y = fmaf(scale, s4.y, x4.y);
        o.z = fmaf(scale, s4.z, x4.z);
        o.w = fmaf(scale, s4.w, x4.w);
        *(float4*)(out + ((size_t)(b * C_CH + c) * P_TOT + tile_p + chunk * 4)) = o;
    }
}

extern "C" void kernel_launch(void* const* d_in, const int* in_sizes, int n_in,
                              void* d_out, int out_size, void* d_ws, size_t ws_size,
                              hipStream_t stream) {
    const float* x1    = (const float*)d_in[0];
    // d_in[1] (x2) cancels out of the reference computation exactly.
    const float* gamma = (const float*)d_in[2];
    float* out = (float*)d_out;

    dim3 grid(16 * (P_TOT / TILE_P));   // 16 batches * 128 p-tiles = 2048 blocks
    gam_fused_kernel<<<grid, 256, 0, stream>>>(x1, gamma, out);
}